// LennardJonesExtension_41790031790530
// MI455X (gfx1250) — compile-verified
//
#include <hip/hip_runtime.h>

// ---------------------------------------------------------------------------
// Lennard-Jones per-atom energy, half neighbor list.
//   Roofline: 168 MB streamed reads / 23.3 TB/s ~= 7.2 us; compute negligible.
//   => optimize data movement only:
//      - 4 pairs/thread, all loads b128, non-temporal (one-shot stream)
//      - scatter via native global_atomic_add_f32 (device scope); 1 MB output
//        stays resident in the 192 MB L2, so atomics never hit HBM.
// ---------------------------------------------------------------------------

typedef float v4f __attribute__((ext_vector_type(4)));
typedef int   v4i __attribute__((ext_vector_type(4)));

// Constants (sigma = eps = 1, cutoff = 2.5), matching the reference exactly.
//   sr_c^6 = (1/2.5)^6 = 64/15625, SHIFT = 4*(sr_c^12 - sr_c^6)
//   half-energy h = 0.5*(4*(sr12 - sr6) - SHIFT) = 2*(sr12 - sr6) - 0.5*SHIFT
__device__ __constant__ double kInv6_d = 64.0 / 15625.0;

__device__ __forceinline__ float half_shift() {
    const double inv6 = 64.0 / 15625.0;
    return (float)(0.5 * (4.0 * (inv6 * inv6 - inv6)));
}

// Fire-and-forget f32 atomic add at device scope: single L2-side RMW,
// tracked by STOREcnt, drained by s_endpgm's implicit wait-idle.
__device__ __forceinline__ void atomic_add_f32_dev(float* p, float v) {
    asm volatile("global_atomic_add_f32 %0, %1, off scope:SCOPE_DEV"
                 :: "v"(p), "v"(v)
                 : "memory");
}

// half-LJ energy for one pair given |r|^2
__device__ __forceinline__ float lj_half(float r2, float hshift) {
    float sr2  = __builtin_amdgcn_rcpf(r2);   // sigma^2 / r^2, sigma = 1
    float sr6  = sr2 * sr2 * sr2;
    float sr12 = sr6 * sr6;
    return 2.0f * (sr12 - sr6) - hshift;      // 0.5 * e_pair
}

__global__ void lj_zero_kernel(float* __restrict__ out, int n) {
    int i = blockIdx.x * blockDim.x + threadIdx.x;
    if (i < n) out[i] = 0.0f;
}

__global__ __launch_bounds__(256)
void lj_pair_kernel(const int* __restrict__ pairs,     // [P,2]
                    const float* __restrict__ dist,    // [P,3]
                    float* __restrict__ out,           // [n_atoms]
                    int npairs) {
    const float hshift = half_shift();

    int g  = blockIdx.x * blockDim.x + threadIdx.x;   // group of 4 pairs
    int p0 = g << 2;
    if (p0 >= npairs) return;

    if (p0 + 4 <= npairs) {
        // ---- fast path: 4 pairs, fully vectorized non-temporal loads ----
        const v4i* pp = (const v4i*)pairs;            // 8 ints  = 2 x b128
        const v4f* dp = (const v4f*)dist;             // 12 flts = 3 x b128
        v4i pa = __builtin_nontemporal_load(pp + 2 * g);
        v4i pb = __builtin_nontemporal_load(pp + 2 * g + 1);
        v4f d0 = __builtin_nontemporal_load(dp + 3 * g);
        v4f d1 = __builtin_nontemporal_load(dp + 3 * g + 1);
        v4f d2 = __builtin_nontemporal_load(dp + 3 * g + 2);

        // pair k displacement components, unpacked from the 12-float stripe
        float r2_0 = d0.x * d0.x + d0.y * d0.y + d0.z * d0.z;
        float r2_1 = d0.w * d0.w + d1.x * d1.x + d1.y * d1.y;
        float r2_2 = d1.z * d1.z + d1.w * d1.w + d2.x * d2.x;
        float r2_3 = d2.y * d2.y + d2.z * d2.z + d2.w * d2.w;

        float h0 = lj_half(r2_0, hshift);
        float h1 = lj_half(r2_1, hshift);
        float h2 = lj_half(r2_2, hshift);
        float h3 = lj_half(r2_3, hshift);

        atomic_add_f32_dev(out + pa.x, h0);
        atomic_add_f32_dev(out + pa.y, h0);
        atomic_add_f32_dev(out + pa.z, h1);
        atomic_add_f32_dev(out + pa.w, h1);
        atomic_add_f32_dev(out + pb.x, h2);
        atomic_add_f32_dev(out + pb.y, h2);
        atomic_add_f32_dev(out + pb.z, h3);
        atomic_add_f32_dev(out + pb.w, h3);
    } else {
        // ---- tail: scalar per-pair (P is a multiple of 4 in practice) ----
        for (int p = p0; p < npairs; ++p) {
            int   i  = pairs[2 * p + 0];
            int   j  = pairs[2 * p + 1];
            float dx = dist[3 * p + 0];
            float dy = dist[3 * p + 1];
            float dz = dist[3 * p + 2];
            float h  = lj_half(dx * dx + dy * dy + dz * dz, hshift);
            atomic_add_f32_dev(out + i, h);
            atomic_add_f32_dev(out + j, h);
        }
    }
}

extern "C" void kernel_launch(void* const* d_in, const int* in_sizes, int n_in,
                              void* d_out, int out_size, void* d_ws, size_t ws_size,
                              hipStream_t stream) {
    const int*   pairs = (const int*)d_in[0];    // [P,2] int32
    const float* dist  = (const float*)d_in[1];  // [P,3] float32
    // d_in[2] = n_atoms (device scalar); out_size == n_atoms, so use that.
    float* out = (float*)d_out;

    const int P = in_sizes[0] / 2;

    // 1) zero the accumulator (d_out is poisoned; atomics accumulate)
    {
        int threads = 256;
        int blocks  = (out_size + threads - 1) / threads;
        lj_zero_kernel<<<blocks, threads, 0, stream>>>(out, out_size);
    }
    // 2) pair kernel: 4 pairs per thread
    {
        int threads = 256;
        int groups  = (P + 3) / 4;
        int blocks  = (groups + threads - 1) / threads;
        lj_pair_kernel<<<blocks, threads, 0, stream>>>(pairs, dist, out, P);
    }
}